// LightGCN_47614007444025
// MI455X (gfx1250) — compile-verified
//
#include <hip/hip_runtime.h>
#include <cstdint>

#define NUSERS 100000
#define NITEMS 50000
#define NNODES (NUSERS + NITEMS)
#define DIM 64
#define NLAYERS 3
#define TILE 32
#define WPB 8        // waves per block
#define TPB 256

// ---- probe: shows atomicAdd(float*) lowering at the top of the disasm ----
__global__ void k_atomic_probe(float* p, float v) { atomicAdd(p, v); }

// ---------------- SpMM with async-LDS edge staging ----------------
// One wave owns a tile of 32 edges. Edge stream (row/col/norm) is staged into
// LDS with global_load_async_to_lds_b32 (double-buffered, ASYNCcnt-tracked).
// Each lane then ds_loads its own staged slot once per tile; the inner loop
// broadcasts edge i's metadata with v_readlane (VALU, no memory traffic).
// Per edge: whole wave processes D=64 floats, float2 per lane -> one 256B
// coalesced gather from h_in and 2 global f32 atomics per lane into h_out.

__global__ void __launch_bounds__(TPB)
k_spmm(const int* __restrict__ row, const int* __restrict__ col,
       const float* __restrict__ nrm, const float* __restrict__ h_in,
       float* __restrict__ h_out, int E, int ntiles) {
  // [wave][buf][field][lane], field: 0=row 1=col 2=norm -> fields 128B apart
  __shared__ unsigned s_edge[WPB][2][3][TILE];

  const int lane = threadIdx.x & 31;
  const int wid  = threadIdx.x >> 5;
  const int wstride = gridDim.x * WPB;
  int tile = blockIdx.x * WPB + wid;
  if (tile >= ntiles) return;

  auto stage = [&](int t, int buf) {
    unsigned e = (unsigned)t * TILE + (unsigned)lane;
    if (e >= (unsigned)E) e = (unsigned)E - 1u;   // clamp tail (dup loads harmless)
    unsigned voff = e * 4u;
    // flat address of a __shared__ object: low 32 bits are the LDS byte offset
    unsigned l0 = (unsigned)(uintptr_t)&s_edge[wid][buf][0][lane];
    unsigned l1 = (unsigned)(uintptr_t)&s_edge[wid][buf][1][lane];
    unsigned l2 = (unsigned)(uintptr_t)&s_edge[wid][buf][2][lane];
    asm volatile("global_load_async_to_lds_b32 %0, %1, %2"
                 :: "v"(l0), "v"(voff), "s"(row) : "memory");
    asm volatile("global_load_async_to_lds_b32 %0, %1, %2"
                 :: "v"(l1), "v"(voff), "s"(col) : "memory");
    asm volatile("global_load_async_to_lds_b32 %0, %1, %2"
                 :: "v"(l2), "v"(voff), "s"(nrm) : "memory");
  };

  stage(tile, 0);
  int buf = 0;
  for (; tile < ntiles; tile += wstride) {
    int nxt = tile + wstride;
    if (nxt < ntiles) {
      stage(nxt, buf ^ 1);                              // prefetch next tile
      asm volatile("s_wait_asynccnt 0x3" ::: "memory"); // current tile's 3 loads done
    } else {
      asm volatile("s_wait_asynccnt 0x0" ::: "memory");
    }

    // lane's own staged slot -> registers (guaranteed DS path, not elidable)
    unsigned laddr = (unsigned)(uintptr_t)&s_edge[wid][buf][0][lane];
    int myrow, mycol, mynrm;
    asm volatile("ds_load_b32 %0, %3\n\t"
                 "ds_load_b32 %1, %3 offset:128\n\t"
                 "ds_load_b32 %2, %3 offset:256\n\t"
                 "s_wait_dscnt 0x0"
                 : "=&v"(myrow), "=&v"(mycol), "=&v"(mynrm)
                 : "v"(laddr) : "memory");

    int cnt = E - tile * TILE;
    if (cnt > TILE) cnt = TILE;
#pragma unroll 4
    for (int i = 0; i < cnt; ++i) {
      // broadcast edge i's metadata from lane i -> SGPRs (v_readlane_b32)
      int   r = __builtin_amdgcn_readlane(myrow, i);
      int   c = __builtin_amdgcn_readlane(mycol, i);
      float w = __int_as_float(__builtin_amdgcn_readlane(mynrm, i));
      const float2 v = *(const float2*)(h_in + (size_t)(unsigned)c * DIM + lane * 2);
      float* dst = h_out + (size_t)(unsigned)r * DIM + lane * 2;
      atomicAdd(dst + 0, v.x * w);
      atomicAdd(dst + 1, v.y * w);
    }
    buf ^= 1;
  }
}

// ---------------- elementwise helpers ----------------

__global__ void k_zero_f4(float4* __restrict__ p, int n4) {
  int i = blockIdx.x * blockDim.x + threadIdx.x;
  int s = gridDim.x * blockDim.x;
  float4 z; z.x = z.y = z.z = z.w = 0.f;
  for (; i < n4; i += s) p[i] = z;
}

__global__ void k_deg(const int* __restrict__ row, float* __restrict__ deg, int E) {
  int i = blockIdx.x * blockDim.x + threadIdx.x;
  int s = gridDim.x * blockDim.x;
  for (; i < E; i += s) atomicAdd(&deg[row[i]], 1.0f);
}

__global__ void k_dinv(const float* __restrict__ deg, float* __restrict__ dinv, int n) {
  int i = blockIdx.x * blockDim.x + threadIdx.x;
  int s = gridDim.x * blockDim.x;
  for (; i < n; i += s) {
    float d = deg[i];
    dinv[i] = (d > 0.f) ? rsqrtf(d) : 0.f;
  }
}

__global__ void k_norm(const int* __restrict__ row, const int* __restrict__ col,
                       const float* __restrict__ dinv, float* __restrict__ nrm, int E) {
  int i = blockIdx.x * blockDim.x + threadIdx.x;
  int s = gridDim.x * blockDim.x;
  for (; i < E; i += s) nrm[i] = dinv[row[i]] * dinv[col[i]];
}

__global__ void k_init(const float4* __restrict__ u, const float4* __restrict__ it,
                       float4* __restrict__ h0, float4* __restrict__ acc,
                       int n4u, int n4tot) {
  int i = blockIdx.x * blockDim.x + threadIdx.x;
  int s = gridDim.x * blockDim.x;
  for (; i < n4tot; i += s) {
    float4 v = (i < n4u) ? u[i] : it[i - n4u];
    h0[i] = v;
    acc[i] = v;
  }
}

__global__ void k_acc(float4* __restrict__ acc, const float4* __restrict__ h, int n4) {
  int i = blockIdx.x * blockDim.x + threadIdx.x;
  int s = gridDim.x * blockDim.x;
  for (; i < n4; i += s) {
    float4 a = acc[i]; float4 b = h[i];
    a.x += b.x; a.y += b.y; a.z += b.z; a.w += b.w;
    acc[i] = a;
  }
}

__global__ void k_scale(float4* __restrict__ p, float c, int n4) {
  int i = blockIdx.x * blockDim.x + threadIdx.x;
  int s = gridDim.x * blockDim.x;
  for (; i < n4; i += s) {
    float4 a = p[i];
    a.x *= c; a.y *= c; a.z *= c; a.w *= c;
    p[i] = a;
  }
}

// ---------------- launcher ----------------

extern "C" void kernel_launch(void* const* d_in, const int* in_sizes, int n_in,
                              void* d_out, int out_size, void* d_ws, size_t ws_size,
                              hipStream_t stream) {
  const float* user = (const float*)d_in[0];
  const float* item = (const float*)d_in[1];
  const int*  edges = (const int*)d_in[2];
  const int E = in_sizes[2] / 2;              // 4,800,000
  const int* row  = edges;
  const int* colp = edges + E;

  // workspace layout (all float, 16B-aligned slices)
  float* deg  = (float*)d_ws;                       // N
  float* dinv = deg  + NNODES;                      // N
  float* nrm  = dinv + NNODES;                      // E
  float* hA   = nrm  + (size_t)E;                   // N*D
  float* hB   = hA   + (size_t)NNODES * DIM;        // N*D
  float* acc  = (float*)d_out;                      // N*D accumulator lives in d_out

  const int n4 = NNODES * DIM / 4;

  k_zero_f4<<<256,  TPB, 0, stream>>>((float4*)deg, NNODES / 4);
  k_deg    <<<4096, TPB, 0, stream>>>(row, deg, E);
  k_dinv   <<<256,  TPB, 0, stream>>>(deg, dinv, NNODES);
  k_norm   <<<4096, TPB, 0, stream>>>(row, colp, dinv, nrm, E);
  k_init   <<<2048, TPB, 0, stream>>>((const float4*)user, (const float4*)item,
                                      (float4*)hA, (float4*)acc,
                                      NUSERS * DIM / 4, n4);

  const int ntiles = (E + TILE - 1) / TILE;
  float* hin = hA;
  float* hout = hB;
  for (int l = 0; l < NLAYERS; ++l) {
    k_zero_f4<<<2048, TPB, 0, stream>>>((float4*)hout, n4);
    k_spmm   <<<2048, TPB, 0, stream>>>(row, colp, nrm, hin, hout, E, ntiles);
    k_acc    <<<2048, TPB, 0, stream>>>((float4*)acc, (const float4*)hout, n4);
    float* t = hin; hin = hout; hout = t;
  }
  k_scale<<<2048, TPB, 0, stream>>>((float4*)acc, 1.0f / (NLAYERS + 1), n4);
}